// GraphStructureEncoder_43439299232177
// MI455X (gfx1250) — compile-verified
//
#include <hip/hip_runtime.h>
#include <hip/hip_bf16.h>

#define GN   50000
#define GE   800000
#define GIN  128
#define GHID 64
#define GH   4
#define NEG_SLOPE 0.2f
#define GEPS 1e-9f

typedef __attribute__((ext_vector_type(2))) float v2f;
typedef __attribute__((ext_vector_type(8))) float v8f;

// ---------------------------------------------------------------------------
// Zero-fill scratch (must be deterministic every call; harness poisons 0xAA)
// ---------------------------------------------------------------------------
__global__ void zero_f32(float* __restrict__ p, int n) {
    int i = blockIdx.x * blockDim.x + threadIdx.x;
    if (i < n) p[i] = 0.0f;
}

// ---------------------------------------------------------------------------
// FP32 GEMM via V_WMMA_F32_16X16X4_F32.
// One wave (32 lanes) computes one 16x16 tile of C = A[M,K] @ B[K,Ncol].
// All dims are multiples of 16 (M=50000=3125*16, K in {128,256}, Ncol in {256,64}),
// so no masking: EXEC stays all-ones inside the WMMA loop as the ISA requires.
// A-layout (ISA 7.12.2, 32-bit A 16x4): lanes 0-15 -> K=0,1 ; lanes 16-31 -> K=2,3.
// ---------------------------------------------------------------------------
__global__ void gemm_wmma_f32(const float* __restrict__ A,
                              const float* __restrict__ B,
                              float* __restrict__ C,
                              int M, int K, int Ncol, int tilesN, int totalTiles) {
    int wave = blockIdx.x * (blockDim.x >> 5) + (threadIdx.x >> 5);
    if (wave >= totalTiles) return;          // wave-uniform exit
    int lane = threadIdx.x & 31;
    int tM = wave / tilesN;
    int tN = wave % tilesN;

    int rowA = tM * 16 + (lane & 15);        // lanes 0-15 and 16-31 both map M=0..15
    int colB = tN * 16 + (lane & 15);
    int khi  = (lane >> 4) << 1;             // 0 for lanes 0-15, 2 for lanes 16-31

    const float* __restrict__ Arow = A + (size_t)rowA * K;

    v8f acc = {};
    for (int k = 0; k < K; k += 4) {
        v2f a, b;
        a.x = Arow[k + khi];
        a.y = Arow[k + khi + 1];
        b.x = B[(size_t)(k + khi)     * Ncol + colB];
        b.y = B[(size_t)(k + khi + 1) * Ncol + colB];
        acc = __builtin_amdgcn_wmma_f32_16x16x4_f32(
            /*neg_a=*/false, a, /*neg_b=*/false, b,
            /*c_mod=*/(short)0, acc, /*reuse_a=*/false, /*reuse_b=*/false);
    }

    // C/D layout: VGPR i -> (M = i + (lane<16 ? 0 : 8), N = lane&15)
    int rbase = tM * 16 + ((lane >> 4) << 3);
    int c     = tN * 16 + (lane & 15);
#pragma unroll
    for (int i = 0; i < 8; ++i)
        C[(size_t)(rbase + i) * Ncol + c] = acc[i];
}

// ---------------------------------------------------------------------------
// el[n,h] = sum_d feat[n,h,d]*al[h,d] ; er likewise. One wave per (n,h) pair,
// wave32 shuffle reduction.
// ---------------------------------------------------------------------------
__global__ void attn_scores(const float* __restrict__ feat,
                            const float* __restrict__ al,
                            const float* __restrict__ ar,
                            float* __restrict__ el,
                            float* __restrict__ er,
                            int NH, int D, int Hh) {
    int wave = blockIdx.x * (blockDim.x >> 5) + (threadIdx.x >> 5);
    if (wave >= NH) return;
    int lane = threadIdx.x & 31;
    int h = wave % Hh;
    const float* __restrict__ f  = feat + (size_t)wave * D;
    const float* __restrict__ vl = al + (size_t)h * D;
    const float* __restrict__ vr = ar + (size_t)h * D;
    float sl = 0.0f, sr = 0.0f;
    for (int d = lane; d < D; d += 32) {
        float v = f[d];
        sl += v * vl[d];
        sr += v * vr[d];
    }
#pragma unroll
    for (int off = 16; off > 0; off >>= 1) {
        sl += __shfl_xor(sl, off, 32);
        sr += __shfl_xor(sr, off, 32);
    }
    if (lane == 0) { el[wave] = sl; er[wave] = sr; }
}

// ---------------------------------------------------------------------------
// Per-edge softmax numerator: e = exp(leaky_relu(el[src]+er[dst])).
// (Max-subtraction is a mathematical no-op for softmax; logits here are tiny.)
// Accumulate normalizer z[dst,h] with f32 atomics (L2-resident).
// ---------------------------------------------------------------------------
__global__ void edge_exp(const int* __restrict__ src, const int* __restrict__ dst,
                         const float* __restrict__ el, const float* __restrict__ er,
                         float* __restrict__ ebuf, float* __restrict__ z,
                         int Ee, int Hh) {
    int i = blockIdx.x * blockDim.x + threadIdx.x;
    if (i >= Ee * Hh) return;
    int e = i / Hh;
    int h = i - e * Hh;
    int s = src[e], d = dst[e];
    float v = el[s * Hh + h] + er[d * Hh + h];
    v = (v > 0.0f) ? v : NEG_SLOPE * v;
    float ex = __expf(v);
    ebuf[i] = ex;
    atomicAdd(&z[d * Hh + h], ex);
}

// ---------------------------------------------------------------------------
// rst[dst,h,d] += (ebuf[e,h]/(z[dst,h]+eps)) * feat[src,h,d]
// HD = H*D values per edge; block of 256 threads covers 256/HD edges.
// ---------------------------------------------------------------------------
__global__ void aggregate(const int* __restrict__ src, const int* __restrict__ dst,
                          const float* __restrict__ ebuf, const float* __restrict__ z,
                          const float* __restrict__ feat, float* __restrict__ rst,
                          int Ee, int Hh, int D, int HD, int edgesPerBlock) {
    int t  = threadIdx.x;
    int le = t / HD;
    int w  = t - le * HD;
    long e = (long)blockIdx.x * edgesPerBlock + le;
    if (e >= Ee) return;
    int h = w / D;
    int s = src[e], d = dst[e];
    float alpha = ebuf[e * Hh + h] / (z[d * Hh + h] + GEPS);
    float val = alpha * feat[(size_t)s * HD + w];
    atomicAdd(&rst[(size_t)d * HD + w], val);
}

// ---------------------------------------------------------------------------
// out[i] = relu(in[i] + bias[i % HD])
// ---------------------------------------------------------------------------
__global__ void bias_relu(const float* __restrict__ in, const float* __restrict__ bias,
                          float* __restrict__ out, int total, int HD) {
    int i = blockIdx.x * blockDim.x + threadIdx.x;
    if (i >= total) return;
    float v = in[i] + bias[i % HD];
    out[i] = (v > 0.0f) ? v : 0.0f;
}

// ---------------------------------------------------------------------------
extern "C" void kernel_launch(void* const* d_in, const int* in_sizes, int n_in,
                              void* d_out, int out_size, void* d_ws, size_t ws_size,
                              hipStream_t stream) {
    // inputs (setup_inputs order): x, src, dst, W1, al1, ar1, b1, W2, al2, ar2, b2
    const float* x   = (const float*)d_in[0];
    const int*   src = (const int*)  d_in[1];
    const int*   dst = (const int*)  d_in[2];
    const float* W1  = (const float*)d_in[3];
    const float* al1 = (const float*)d_in[4];
    const float* ar1 = (const float*)d_in[5];
    const float* b1  = (const float*)d_in[6];
    const float* W2  = (const float*)d_in[7];
    const float* al2 = (const float*)d_in[8];
    const float* ar2 = (const float*)d_in[9];
    const float* b2  = (const float*)d_in[10];
    float* out = (float*)d_out;

    // workspace layout (floats)
    float* ws = (float*)d_ws;
    size_t o = 0;
    float* feat1 = ws + o; o += (size_t)GN * GH * GHID;   // 12.8M
    float* rst1  = ws + o; o += (size_t)GN * GH * GHID;   // 12.8M (also h1 after bias+relu)
    float* feat2 = ws + o; o += (size_t)GN * GHID;        // 3.2M
    float* rst2  = ws + o; o += (size_t)GN * GHID;        // 3.2M
    float* el1   = ws + o; o += (size_t)GN * GH;
    float* er1   = ws + o; o += (size_t)GN * GH;
    float* z1    = ws + o; o += (size_t)GN * GH;
    float* ebuf1 = ws + o; o += (size_t)GE * GH;          // 3.2M
    float* el2   = ws + o; o += (size_t)GN;
    float* er2   = ws + o; o += (size_t)GN;
    float* z2    = ws + o; o += (size_t)GN;
    float* ebuf2 = ws + o; o += (size_t)GE;

    const int BLK = 256;
    auto cdiv = [](int a, int b) { return (a + b - 1) / b; };

    // ---- zero accumulators (every call; graph-capture safe) ----
    zero_f32<<<cdiv(GN * GH * GHID, BLK), BLK, 0, stream>>>(rst1, GN * GH * GHID);
    zero_f32<<<cdiv(GN * GH,        BLK), BLK, 0, stream>>>(z1,   GN * GH);
    zero_f32<<<cdiv(GN * GHID,      BLK), BLK, 0, stream>>>(rst2, GN * GHID);
    zero_f32<<<cdiv(GN,             BLK), BLK, 0, stream>>>(z2,   GN);

    // ---- layer 1 ----
    {   // feat1[N,256] = x[N,128] @ W1[128,256]
        int tilesN = (GH * GHID) / 16;            // 16
        int totalTiles = (GN / 16) * tilesN;      // 50000
        int blocks = cdiv(totalTiles, BLK / 32);  // 8 waves/block
        gemm_wmma_f32<<<blocks, BLK, 0, stream>>>(x, W1, feat1, GN, GIN, GH * GHID,
                                                  tilesN, totalTiles);
    }
    attn_scores<<<cdiv(GN * GH, BLK / 32), BLK, 0, stream>>>(feat1, al1, ar1, el1, er1,
                                                             GN * GH, GHID, GH);
    edge_exp<<<cdiv(GE * GH, BLK), BLK, 0, stream>>>(src, dst, el1, er1, ebuf1, z1, GE, GH);
    {   // 256 values per edge -> 1 edge per 256-thread block
        int HD = GH * GHID;                       // 256
        int epb = BLK / HD;                       // 1
        aggregate<<<cdiv(GE, epb), BLK, 0, stream>>>(src, dst, ebuf1, z1, feat1, rst1,
                                                     GE, GH, GHID, HD, epb);
    }
    bias_relu<<<cdiv(GN * GH * GHID, BLK), BLK, 0, stream>>>(rst1, b1, rst1,
                                                             GN * GH * GHID, GH * GHID);

    // ---- layer 2 ----
    {   // feat2[N,64] = h1[N,256] @ W2[256,64]
        int tilesN = GHID / 16;                   // 4
        int totalTiles = (GN / 16) * tilesN;      // 12500
        int blocks = cdiv(totalTiles, BLK / 32);
        gemm_wmma_f32<<<blocks, BLK, 0, stream>>>(rst1, W2, feat2, GN, GH * GHID, GHID,
                                                  tilesN, totalTiles);
    }
    attn_scores<<<cdiv(GN, BLK / 32), BLK, 0, stream>>>(feat2, al2, ar2, el2, er2,
                                                        GN, GHID, 1);
    edge_exp<<<cdiv(GE, BLK), BLK, 0, stream>>>(src, dst, el2, er2, ebuf2, z2, GE, 1);
    {   // 64 values per edge -> 4 edges per block
        int HD = GHID;                            // 64
        int epb = BLK / HD;                       // 4
        aggregate<<<cdiv(GE, epb), BLK, 0, stream>>>(src, dst, ebuf2, z2, feat2, rst2,
                                                     GE, 1, GHID, HD, epb);
    }
    bias_relu<<<cdiv(GN * GHID, BLK), BLK, 0, stream>>>(rst2, b2, out, GN * GHID, GHID);
}